// AttentionBlock_1752346657385
// MI455X (gfx1250) — compile-verified
//
#include <hip/hip_runtime.h>

// Problem constants
#define B_  4
#define C_  64
#define H_  96
#define W_  96
#define SP  (H_ * W_)      // 9216 spatial per plane
#define PH  102            // padded plane edge (96 + 6)
#define PSP (PH * PH)      // 10404

typedef __attribute__((ext_vector_type(2))) float v2f;
typedef __attribute__((ext_vector_type(8))) float v8f;

// ---------------------------------------------------------------------------
// Kernel 1: fused q/k/v 1x1 conv as GEMM, fp32 WMMA 16x16x4.
//   ws layout: ws[(sel*4 + b)*64 + c][9216], sel: 0=q 1=k 2=v  (28.3 MB)
//   out[o][n] = sum_c W[o][c] * X[c][n] + bias[o]
// One wave computes one 16(M=out ch) x 16(N=spatial) tile, K=64 in 16 steps.
// ---------------------------------------------------------------------------
__global__ __launch_bounds__(128) void qkv_conv_wmma(
    const float* __restrict__ x,
    const float* __restrict__ qw, const float* __restrict__ qb,
    const float* __restrict__ kw, const float* __restrict__ kb,
    const float* __restrict__ vw, const float* __restrict__ vb,
    float* __restrict__ ws)
{
  const int lane  = threadIdx.x & 31;
  const int wave  = threadIdx.x >> 5;
  const int ntile = blockIdx.x * 4 + wave;   // 0..575
  const int n0    = ntile * 16;
  const int mtile = blockIdx.y;              // 0..11
  const int sel   = mtile >> 2;              // 0=q 1=k 2=v (16-tiles never straddle)
  const int m0    = (mtile & 3) * 16;        // out-channel base within 64
  const int b     = blockIdx.z;

  const float* Wm; const float* bias;
  if (sel == 0)      { Wm = qw; bias = qb; }
  else if (sel == 1) { Wm = kw; bias = kb; }
  else               { Wm = vw; bias = vb; }

  const float* X   = x + (size_t)b * C_ * SP;
  const int l15    = lane & 15;
  const int khalf  = (lane >> 4) * 2;        // K offset for this lane half

  // A(16x4): lane holds row M=m0+l15, VGPR0=K(khalf), VGPR1=K(khalf+1)
  // B(4x16): lane holds col N=n0+l15, same K mapping
  const float* wrow = Wm + (m0 + l15) * C_ + khalf;  // 8B-aligned pairs
  const float* xcol = X + n0 + l15;

  v8f acc = {0.f, 0.f, 0.f, 0.f, 0.f, 0.f, 0.f, 0.f};
#pragma unroll
  for (int k0 = 0; k0 < C_; k0 += 4) {
    // A pair as one b64 load (elements k0+khalf, k0+khalf+1)
    v2f a = *(const v2f*)(wrow + k0);
    v2f bm;
    bm.x = xcol[(size_t)(k0 + khalf) * SP];
    bm.y = xcol[(size_t)(k0 + khalf + 1) * SP];
    // (neg_a, A, neg_b, B, c_mod, C, reuse_a, reuse_b)
    acc = __builtin_amdgcn_wmma_f32_16x16x4_f32(false, a, false, bm,
                                                (short)0, acc, false, false);
  }

  // C/D layout: VGPR r -> M = m0 + r + 8*(lane>=16), N = n0 + (lane&15)
  float* outb  = ws + ((size_t)(sel * B_ + b) * C_) * SP + n0 + l15;
  const int mh = (lane >> 4) * 8;
#pragma unroll
  for (int r = 0; r < 8; ++r) {
    int ch = m0 + r + mh;
    outb[(size_t)ch * SP] = acc[r] + bias[ch];
  }
}

// ---------------------------------------------------------------------------
// Kernel 2: per-(b,c) local attention, planes staged in LDS.
//   Site (x, g): q = qconv[x, 16g+d], d=0..15
//   qk[i,j] = sum_d q[d] * kpad[x+i, 16g+j+d] + qsum*rel[i,j]
//   softmax over 49, attn = sum w * vpad window, ReLU.
//   kpad/vpad halo (conv of zero padding) == bias.
// 576 threads: tid -> g = tid/96 (wave-uniform), x = tid%96.
// ---------------------------------------------------------------------------
__global__ __launch_bounds__(576) void attn_kernel(
    const float* __restrict__ ws,
    const float* __restrict__ kb, const float* __restrict__ vb,
    const float* __restrict__ rel_h, const float* __restrict__ rel_w,
    float* __restrict__ out)
{
  extern __shared__ float smem[];
  float* kL   = smem;                 // PSP floats (padded coords)
  float* vL   = smem + PSP;           // PSP
  float* qL   = smem + 2 * PSP;       // SP (byte offset 83232, 16B aligned)
  float* relL = smem + 2 * PSP + SP;  // 49 (+pad)

  const int c   = blockIdx.x;
  const int b   = blockIdx.y;
  const int tid = threadIdx.x;

  const float* qsrc = ws + ((size_t)(0 * B_ + b) * C_ + c) * SP;
  const float* ksrc = ws + ((size_t)(1 * B_ + b) * C_ + c) * SP;
  const float* vsrc = ws + ((size_t)(2 * B_ + b) * C_ + c) * SP;
  const float kbias = kb[c];
  const float vbias = vb[c];

  // Pull the three planes toward the WGP: 576 lanes * stride-16 floats
  // covers every 64B chunk of each 36KB plane once (global_prefetch_b8).
  {
    int pf = tid * 16;               // [0, 9216) exactly
    __builtin_prefetch(ksrc + pf, 0, 0);
    __builtin_prefetch(vsrc + pf, 0, 0);
    __builtin_prefetch(qsrc + pf, 0, 0);
  }

  // rel[i,j] = sum_c rel_h[c][i][0] + sum_c rel_w[c][0][j]
  if (tid < 49) {
    int i = tid / 7, j = tid - (tid / 7) * 7;
    float r = 0.f;
    for (int cc = 0; cc < 32; ++cc) r += rel_h[cc * 7 + i] + rel_w[cc * 7 + j];
    relL[tid] = r;
  }

  // Stage k/v with bias halo.
  for (int t = tid; t < PSP; t += 576) {
    int px = t / PH, py = t - px * PH;
    int ox = px - 3, oy = py - 3;
    bool inb = ((unsigned)ox < (unsigned)H_) && ((unsigned)oy < (unsigned)W_);
    int s = ox * W_ + oy;
    kL[t] = inb ? ksrc[s] : kbias;
    vL[t] = inb ? vsrc[s] : vbias;
  }
  // Stage q with b128 copies: 9216 floats = 2304 float4, 4 passes of 576.
  for (int t = tid * 4; t < SP; t += 576 * 4) {
    *(float4*)(qL + t) = *(const float4*)(qsrc + t);
  }
  __syncthreads();

  const int g  = tid / 96;       // 0..5, uniform per wave (96 = 3*32)
  const int xx = tid - g * 96;   // 0..95
  const int yb = g * 16;

  float qreg[16];
  float qsum = 0.f;
#pragma unroll
  for (int d = 0; d < 16; ++d) {
    qreg[d] = qL[xx * W_ + yb + d];
    qsum   += qreg[d];
  }

  float p[49];
  float mx = -3.0e38f;
#pragma unroll
  for (int i = 0; i < 7; ++i) {
#pragma unroll
    for (int j = 0; j < 7; ++j) {
      const float* kr = kL + (xx + i) * PH + yb + j;
      float s2 = qsum * relL[i * 7 + j];
#pragma unroll
      for (int d = 0; d < 16; ++d) s2 = __builtin_fmaf(qreg[d], kr[d], s2);
      p[i * 7 + j] = s2;
      mx = fmaxf(mx, s2);
    }
  }

  float denom = 0.f;
#pragma unroll
  for (int kk = 0; kk < 49; ++kk) {
    float e = __expf(p[kk] - mx);
    p[kk] = e;
    denom += e;
  }
  const float inv = 1.0f / denom;

  float o[16];
#pragma unroll
  for (int d = 0; d < 16; ++d) o[d] = 0.f;
#pragma unroll
  for (int i = 0; i < 7; ++i) {
#pragma unroll
    for (int j = 0; j < 7; ++j) {
      float wgt = p[i * 7 + j];
      const float* vr = vL + (xx + i) * PH + yb + j;
#pragma unroll
      for (int d = 0; d < 16; ++d) o[d] = __builtin_fmaf(wgt, vr[d], o[d]);
    }
  }

  float* ob = out + (((size_t)b * C_ + c) * H_ + xx) * W_ + yb;
#pragma unroll
  for (int d = 0; d < 16; ++d) ob[d] = fmaxf(o[d] * inv, 0.f);
}

// ---------------------------------------------------------------------------
extern "C" void kernel_launch(void* const* d_in, const int* in_sizes, int n_in,
                              void* d_out, int out_size, void* d_ws, size_t ws_size,
                              hipStream_t stream) {
  (void)in_sizes; (void)n_in; (void)out_size; (void)ws_size;
  const float* x  = (const float*)d_in[0];
  const float* qw = (const float*)d_in[1];
  const float* qb = (const float*)d_in[2];
  const float* kw = (const float*)d_in[3];
  const float* kb = (const float*)d_in[4];
  const float* vw = (const float*)d_in[5];
  const float* vb = (const float*)d_in[6];
  const float* rh = (const float*)d_in[7];
  const float* rw = (const float*)d_in[8];
  float* out = (float*)d_out;
  float* ws  = (float*)d_ws;   // needs 3*4*64*9216*4 B = 28.3 MB scratch

  // Kernel 1: 576 N-tiles x 12 M-tiles x 4 batches; 4 waves (tiles) per block.
  dim3 g1(144, 12, 4);
  qkv_conv_wmma<<<g1, 128, 0, stream>>>(x, qw, qb, kw, kb, vw, vb, ws);

  // Kernel 2: one block per (c, b); ~117.5 KB dynamic LDS (<< 320 KB/WGP).
  size_t smem = (size_t)(2 * PSP + SP + 64) * sizeof(float);
  dim3 g2(C_, B_);
  attn_kernel<<<g2, 576, smem, stream>>>(ws, kb, vb, rh, rw, out);
}